// MICLayer_58755152610028
// MI455X (gfx1250) — compile-verified
//
#include <hip/hip_runtime.h>

// QAM-64 vector quantization, forward = hard nearest-codeword (straight-through).
// score = s.c - 0.5*|c|^2 via V_WMMA_F32_16X16X4_F32 with A = codebook, B = symbols,
// so D is (entries x symbols): argmax is in-lane (max-tree + independent equality
// scan) plus a single xor-16 lane-pair exchange. Branchless throughout.

typedef __attribute__((ext_vector_type(2))) float v2f;
typedef __attribute__((ext_vector_type(8))) float v8f;

#define PLANE 1024          // H*W = 32*32
#define CLIPV 2.0f

__global__ __launch_bounds__(256)
void qam_vq_wmma(const float* __restrict__ z,
                 const float* __restrict__ cb,   // [64][2] raw codebook
                 float* __restrict__ out,
                 int numSuper)                   // super-tiles of 32 symbols
{
    const int lane = threadIdx.x & 31;
    const int lp   = lane & 15;          // symbol position within tile (N)
    const int half = lane >> 4;          // 0: entries M=0..7 / I-writer, 1: M=8..15 / Q-writer

    const int wavesPerBlk = blockDim.x >> 5;
    const int gwave  = blockIdx.x * wavesPerBlk + (threadIdx.x >> 5);
    const int nwaves = gridDim.x * wavesPerBlk;

    // ---- codebook normalization: power = mean(|c|^2) over 64 entries ----
    float4 c4 = ((const float4*)cb)[lane];            // entries 2*lane, 2*lane+1
    float ss = c4.x*c4.x + c4.y*c4.y + c4.z*c4.z + c4.w*c4.w;
    #pragma unroll
    for (int m = 16; m >= 1; m >>= 1) ss += __shfl_xor(ss, m, 32);
    const float power = ss * (1.0f / 64.0f);
    const float scale = sqrtf(1.0f / (power + 1e-8f));
    const float nhalf = -0.5f * scale * scale;

    // ---- A operands: codebook groups g (16 entries x K=4, MxK layout) ----
    // lanes 0-15 (entry M=lp): v0 = K0 = c_i, v1 = K1 = c_q
    // lanes 16-31 (entry M=lp): v0 = K2 = -0.5|c|^2, v1 = K3 = 0
    v2f Ag[4];
    #pragma unroll
    for (int g = 0; g < 4; ++g) {
        const int e = g * 16 + lp;
        const float ci = cb[2*e], cq = cb[2*e + 1];
        const float ht = nhalf * (ci*ci + cq*cq);
        Ag[g][0] = half ? ht   : ci * scale;
        Ag[g][1] = half ? 0.0f : cq * scale;
    }

    for (int t = gwave; t < numSuper; t += nwaves) {
        // 32 consecutive symbols inside one H*W plane (1024 % 32 == 0)
        const int p = t >> 5;            // plane index
        const int o = (t & 31) << 5;     // offset in plane
        const float* zi = z + (size_t)p * (2 * PLANE) + o;   // I plane
        const float* zq = zi + PLANE;                        // Q plane
        float* ob = out + (size_t)p * (2 * PLANE) + o;

        #pragma unroll
        for (int sub = 0; sub < 2; ++sub) {
            const int so = sub * 16;

            // B operand: symbols (K=4 x N=16). All lanes load (coalesced),
            // selects keep EXEC uniform; lanes 16-31 carry bias rows K2=1, K3=0.
            float si = zi[so + lp], sq = zq[so + lp];
            si = fminf(fmaxf(si, -CLIPV), CLIPV);
            sq = fminf(fmaxf(sq, -CLIPV), CLIPV);
            v2f B;
            B[0] = half ? 1.0f : si;
            B[1] = half ? 0.0f : sq;

            v8f D[4];
            #pragma unroll
            for (int g = 0; g < 4; ++g) {
                v8f cz = {};
                D[g] = __builtin_amdgcn_wmma_f32_16x16x4_f32(
                           false, Ag[g], false, B, (short)0, cz, false, false);
            }

            // (1) value max over the 32 in-lane scores: balanced tree
            //     (log-depth, max3/dual-issue friendly).
            float gm[4];
            #pragma unroll
            for (int g = 0; g < 4; ++g) {
                const float a = fmaxf(fmaxf(D[g][0], D[g][1]), fmaxf(D[g][2], D[g][3]));
                const float b = fmaxf(fmaxf(D[g][4], D[g][5]), fmaxf(D[g][6], D[g][7]));
                gm[g] = fmaxf(a, b);
            }
            const float bs = fmaxf(fmaxf(gm[0], gm[1]), fmaxf(gm[2], gm[3]));

            // (2) slot recovery: descending equality scan -> lowest matching slot.
            //     All compares independent (ILP); cndmask chain only.
            //     slot = g*8 + r; ascending slot == ascending codebook index here.
            int bslot = 31;
            #pragma unroll
            for (int slot = 30; slot >= 0; --slot) {
                bslot = (D[slot >> 3][slot & 7] == bs) ? slot : bslot;
            }
            // codebook index: entry M = r + 8*half within group g
            int bi = ((bslot >> 3) << 4) + (bslot & 7) + (half << 3);

            // Combine with partner lane (other 32 entries of the same symbol).
            // Bitwise ops (no short-circuit) => branchless cndmask selection.
            const float os  = __shfl_xor(bs, 16, 32);
            const int   oi2 = __shfl_xor(bi, 16, 32);
            const int take = (int)(os > bs) | ((int)(os == bs) & (int)(oi2 < bi));
            bi = take ? oi2 : bi;

            // lanes 0-15 store I components (64B contiguous), 16-31 store Q.
            const float val = cb[2 * bi + half] * scale;
            ob[half * PLANE + so + lp] = val;
        }
    }
}

extern "C" void kernel_launch(void* const* d_in, const int* in_sizes, int n_in,
                              void* d_out, int out_size, void* d_ws, size_t ws_size,
                              hipStream_t stream) {
    const float* z  = (const float*)d_in[0];   // [8,256,32,32] f32
    const float* cb = (const float*)d_in[1];   // [64,2] f32
    float* out = (float*)d_out;

    const int nsym     = in_sizes[0] / 2;      // 1,048,576 symbols
    const int numSuper = nsym / 32;            // 32,768 super-tiles

    dim3 block(256);                           // 8 waves per block
    dim3 grid(2048);                           // grid-stride over super-tiles
    qam_vq_wmma<<<grid, block, 0, stream>>>(z, cb, out, numSuper);
}